// PoseNet_43688407335600
// MI455X (gfx1250) — compile-verified
//
#include <hip/hip_runtime.h>

// cost = 0.5 * sum_n x_n^T C_n x_n, 8192 blocks of 64x64 f32.
// HBM-bound (128 MB of C read once). WMMA f32 16x16x4 does the block matvec
// with x broadcast across B's 16 columns; redundancy factor 16 folded into
// the final scale (0.5/16).

#define N_BLOCKS 8192
#define BLK      64
#define NWG      512   // workgroups in stage 1
#define WGS      256   // threads per workgroup (8 wave32s)

typedef __attribute__((ext_vector_type(2))) float v2f;
typedef __attribute__((ext_vector_type(4))) float v4f;
typedef __attribute__((ext_vector_type(8))) float v8f;

#if defined(__has_builtin)
#  if __has_builtin(__builtin_amdgcn_wmma_f32_16x16x4_f32)
#    define HAVE_WMMA_F32X4 1
#  endif
#endif

__global__ __launch_bounds__(WGS) void quadform_wmma(
    const float* __restrict__ x, const float* __restrict__ C,
    float* __restrict__ ws) {
  const int lane = threadIdx.x & 31;
  const int wave = blockIdx.x * (WGS >> 5) + (threadIdx.x >> 5);
  const int m = lane & 15;        // row within 16-row stripe / column id
  const int h = lane >> 4;        // K-half select (0 or 1)
  const int nWaves  = NWG * (WGS >> 5);       // 4096
  const int perWave = N_BLOCKS / nWaves;      // 2

  float partial = 0.0f;

  for (int bi = 0; bi < perWave; ++bi) {
    const int blk = wave * perWave + bi;
    const float* __restrict__ Cb = C + (size_t)blk * (BLK * BLK);
    const float* __restrict__ xb = x + blk * BLK;

    // Warm caches for the next block while this one computes (uniform branch).
    if (bi + 1 < perWave)
      __builtin_prefetch(Cb + BLK * BLK + lane * 128, 0, 0);

#if HAVE_WMMA_F32X4
    #pragma unroll
    for (int i = 0; i < 4; ++i) {            // four 16-row stripes
      v8f acc = {};
      const float* Ca = Cb + (16 * i + m) * BLK + 2 * h;  // A: 16x4 f32 tile
      const float* Bx = xb + 2 * h;                       // B: x broadcast
      #pragma unroll
      for (int kc = 0; kc < 16; ++kc) {      // K = 64 in steps of 4
        v2f a = *(const v2f*)(Ca + 4 * kc);
        v2f b = *(const v2f*)(Bx + 4 * kc);
        // D = A*B + D ; emits v_wmma_f32_16x16x4_f32
        acc = __builtin_amdgcn_wmma_f32_16x16x4_f32(
            false, a, false, b, (short)0, acc, false, false);
      }
      // acc[r] (lane L) holds y[16i + r + 8h]; weight by matching x, sum.
      const float* xw = xb + 16 * i + 8 * h;
      v4f w0 = *(const v4f*)(xw);
      v4f w1 = *(const v4f*)(xw + 4);
      partial += acc[0] * w0.x + acc[1] * w0.y + acc[2] * w0.z + acc[3] * w0.w
               + acc[4] * w1.x + acc[5] * w1.y + acc[6] * w1.z + acc[7] * w1.w;
    }
#else
    // VALU fallback (kept numerically consistent with the WMMA path's x16
    // redundancy so the same final scale applies).
    for (int rr = 0; rr < 2; ++rr) {
      const int row = lane + 32 * rr;
      float y = 0.0f;
      for (int c = 0; c < BLK; ++c) y += Cb[row * BLK + c] * xb[c];
      partial += 16.0f * xb[row] * y;
    }
#endif
  }

  // Deterministic workgroup reduction -> one float per workgroup.
  __shared__ float sred[WGS];
  sred[threadIdx.x] = partial;
  __syncthreads();
  for (int o = WGS >> 1; o > 0; o >>= 1) {
    if (threadIdx.x < o) sred[threadIdx.x] += sred[threadIdx.x + o];
    __syncthreads();
  }
  if (threadIdx.x == 0) ws[blockIdx.x] = sred[0];
}

__global__ __launch_bounds__(256) void quadform_reduce(
    const float* __restrict__ ws, float* __restrict__ out) {
  __shared__ float s[256];
  const int t = threadIdx.x;
  s[t] = ws[t] + ws[t + 256];   // NWG == 512
  __syncthreads();
  for (int o = 128; o > 0; o >>= 1) {
    if (t < o) s[t] += s[t + o];
    __syncthreads();
  }
  // 0.5 (reference) * 1/16 (WMMA column redundancy)
  if (t == 0) out[0] = s[0] * (0.5f / 16.0f);
}

extern "C" void kernel_launch(void* const* d_in, const int* in_sizes, int n_in,
                              void* d_out, int out_size, void* d_ws, size_t ws_size,
                              hipStream_t stream) {
  const float* x = (const float*)d_in[0];   // [524288] f32
  const float* C = (const float*)d_in[1];   // [8192,64,64] f32
  float* ws = (float*)d_ws;                 // >= NWG floats
  quadform_wmma<<<NWG, WGS, 0, stream>>>(x, C, ws);
  quadform_reduce<<<1, 256, 0, stream>>>(ws, (float*)d_out);
}